// Class_head_MetaOpt_32744830665382
// MI455X (gfx1250) — compile-verified
//
#include <hip/hip_runtime.h>
#include <hip/hip_bf16.h>
#include <math.h>

typedef float v2f __attribute__((ext_vector_type(2)));
typedef float v8f __attribute__((ext_vector_type(8)));

#define N_WAY   10
#define N_SHOT  15
#define N_SUP   150
#define N_QUERY 8192
#define D_FEAT  1024
#define NTOT    1500      // n_sup * n_way
#define KLD     160       // padded leading dim for K
#define HLD     152       // padded leading dim for 150x150 matrices
#define MAX_ITER 15

// ---- workspace layout (float offsets); total ~533K floats = ~2.14 MB ----
#define OFF_K     0         // 160*160 gram (only [0..149]^2 valid)
#define OFF_Z     25600     // 1500 (+pad)
#define OFF_S     27104
#define OFF_LAM   28608
#define OFF_NU    30112     // 150 (+pad)
#define OFF_R1    30272
#define OFF_HR1   31776
#define OFF_RP    33280
#define OFF_RC    33440
#define OFF_SMAT  34944     // 150*152 Schur
#define OFF_RHS   57744
#define OFF_DNU   57904
#define OFF_DZ    58064
#define OFF_DLAM  59568
#define OFF_ALPHA 61072     // [0] = min step-ratio as uint bits
#define OFF_HBUF  61088     // 10 * 150*152 per-way H / chol factor
#define OFF_G     289088    // 10 * 150*152 per-way H^-1
#define MAT_STRIDE 22800
#define OFF_W     517088    // 1024*16 (W = S^T z, padded to 16 cols)

// =====================================================================
// K = S S^T via V_WMMA_F32_16X16X4_F32. One wave per 16x16 output tile.
// A 16x4 f32: lanes 0-15 M=0..15; VGPR0:K=2*hi, VGPR1:K=2*hi+1.
// B 4x16: VGPR0 row K=2*hi, VGPR1 row K=2*hi+1, col = lane&15.
// C/D: VGPR v -> row v+8*hi, col lane&15.
// =====================================================================
__global__ __launch_bounds__(32)
void k_gram(const float* __restrict__ feat, float* __restrict__ Kmat) {
  const int lane = threadIdx.x;
  const int mh = lane & 15, hi = lane >> 4;
  const int r0 = blockIdx.y * 16, c0 = blockIdx.x * 16;
  const float* Ar = feat + (size_t)(r0 + mh) * D_FEAT;
  const float* Br = feat + (size_t)(c0 + mh) * D_FEAT;   // B[k][n] = S[c0+n][k]
  v8f acc = {};
  for (int k0 = 0; k0 < D_FEAT; k0 += 4) {
    v2f a, b;
    a.x = Ar[k0 + 2 * hi];     a.y = Ar[k0 + 2 * hi + 1];
    b.x = Br[k0 + 2 * hi];     b.y = Br[k0 + 2 * hi + 1];
    acc = __builtin_amdgcn_wmma_f32_16x16x4_f32(false, a, false, b,
                                                (short)0, acc, false, false);
  }
  for (int v = 0; v < 8; ++v)
    Kmat[(r0 + v + 8 * hi) * KLD + c0 + mh] = acc[v];
}

// =====================================================================
// In-place blocked Cholesky (lower), one workgroup.
// =====================================================================
__device__ void block_cholesky(float* A, int n, int ld, int tid, int nt) {
  for (int k = 0; k < n; ++k) {
    if (tid == 0) A[k * ld + k] = sqrtf(A[k * ld + k]);
    __syncthreads();
    float dk = A[k * ld + k];
    for (int i = k + 1 + tid; i < n; i += nt) A[i * ld + k] /= dk;
    __syncthreads();
    int t = n - k - 1;
    for (int idx = tid; idx < t * t; idx += nt) {
      int i = k + 1 + idx / t, j = k + 1 + idx % t;
      if (j <= i) A[i * ld + j] -= A[i * ld + k] * A[j * ld + k];
    }
    __syncthreads();
  }
}

// =====================================================================
// QP init: z = h - 1 = 0.1*onehot - 1 ; s = lam = 1 ; nu = 0
// =====================================================================
__global__ __launch_bounds__(512)
void k_qp_init(float* __restrict__ ws, const int* __restrict__ lbl_sup) {
  const int tid = threadIdx.x, nt = 512;
  float* zv = ws + OFF_Z; float* sv = ws + OFF_S; float* lv = ws + OFF_LAM;
  for (int t = tid; t < NTOT; t += nt) {
    int i = t / N_WAY, a = t % N_WAY;
    float oh = (lbl_sup[i / N_SHOT] == a) ? 1.f : 0.f;
    zv[t] = 0.1f * oh - 1.f;
    sv[t] = 1.f;  lv[t] = 1.f;
  }
  for (int i = tid; i < N_SUP; i += nt) (ws + OFF_NU)[i] = 0.f;
}

// =====================================================================
// Per-iteration residuals: mu, rc, r1, rp; reset step-ratio atomic slot.
// =====================================================================
__global__ __launch_bounds__(512)
void k_resid(float* __restrict__ ws, const int* __restrict__ lbl_sup) {
  const int tid = threadIdx.x, nt = 512;
  const float* Kmat = ws + OFF_K;
  const float* zv = ws + OFF_Z; const float* sv = ws + OFF_S;
  const float* lv = ws + OFF_LAM; const float* nuv = ws + OFF_NU;
  float* r1 = ws + OFF_R1; float* rp = ws + OFF_RP; float* rc = ws + OFF_RC;
  __shared__ float sred[512];
  __shared__ float s_mu;

  float part = 0.f;
  for (int t = tid; t < NTOT; t += nt) part += sv[t] * lv[t];
  sred[tid] = part; __syncthreads();
  for (int w = nt / 2; w > 0; w >>= 1) {
    if (tid < w) sred[tid] += sred[tid + w];
    __syncthreads();
  }
  if (tid == 0) s_mu = sred[0] / (float)NTOT;
  __syncthreads();
  float mu = s_mu;

  for (int t = tid; t < NTOT; t += nt) {
    int i = t / N_WAY, a = t % N_WAY;
    float qz = 10.f * zv[t];
    const float* Krow = Kmat + i * KLD;
    float q0 = 0.f, q1 = 0.f, q2 = 0.f, q3 = 0.f;
    for (int j = 0; j < N_SUP; j += 2) {          // 150 even
      q0 += Krow[j]     * zv[j * N_WAY + a];
      q1 += Krow[j + 1] * zv[(j + 1) * N_WAY + a];
    }
    qz += (q0 + q1) + (q2 + q3);
    float oh  = (lbl_sup[i / N_SHOT] == a) ? 1.f : 0.f;   // p = -onehot
    float rd  = qz - oh + lv[t] + nuv[i];
    float rct = sv[t] * lv[t] - 0.1f * mu;
    rc[t] = rct;
    r1[t] = -rd + rct / sv[t];
  }
  for (int i = tid; i < N_SUP; i += nt) {          // rp = A z
    float acc = 0.f;
    for (int a = 0; a < N_WAY; ++a) acc += zv[i * N_WAY + a];
    rp[i] = acc;
  }
  if (tid == 0)
    ((unsigned int*)(ws + OFF_ALPHA))[0] = __float_as_uint(1e30f);
}

// =====================================================================
// Per-way (blockIdx.x = way): build H_a = K + (10 + lam/s) I, Cholesky,
// solve 151 RHS (identity -> G_a = H_a^-1 ; r1_a -> Hinv_r1 column).
// 4-way partial sums break the dependent FMA chains.
// =====================================================================
__global__ __launch_bounds__(256)
void k_way(float* __restrict__ ws) {
  const int tid = threadIdx.x, nt = 256;
  const int a = blockIdx.x;
  const float* Kmat = ws + OFF_K;
  const float* lv = ws + OFF_LAM; const float* sv = ws + OFF_S;
  const float* r1 = ws + OFF_R1;
  float* hr1 = ws + OFF_HR1;
  float* H  = ws + OFF_HBUF + a * MAT_STRIDE;
  float* Ga = ws + OFF_G    + a * MAT_STRIDE;

  for (int idx = tid; idx < N_SUP * N_SUP; idx += nt) {
    int i = idx / N_SUP, j = idx % N_SUP;
    float v = Kmat[i * KLD + j];
    if (i == j) v += 10.f + lv[i * N_WAY + a] / sv[i * N_WAY + a];
    H[i * HLD + j] = v;
  }
  __syncthreads();
  block_cholesky(H, N_SUP, HLD, tid, nt);

  if (tid < N_SUP + 1) {
    const int c = tid;
    float* col; int cs; float b0;
    if (c < N_SUP) { col = Ga + c;  cs = HLD; }
    else           { col = hr1 + a; cs = N_WAY; }
    // forward: L y = b
    for (int r = 0; r < N_SUP; ++r) {
      if (c < N_SUP) b0 = (r == c) ? 1.f : 0.f;
      else           b0 = r1[r * N_WAY + a];
      const float* Hr = H + r * HLD;
      float s0 = 0.f, s1 = 0.f, s2 = 0.f, s3 = 0.f;
      int q = 0;
      for (; q + 3 < r; q += 4) {
        s0 += Hr[q]     * col[q * cs];
        s1 += Hr[q + 1] * col[(q + 1) * cs];
        s2 += Hr[q + 2] * col[(q + 2) * cs];
        s3 += Hr[q + 3] * col[(q + 3) * cs];
      }
      for (; q < r; ++q) s0 += Hr[q] * col[q * cs];
      col[r * cs] = (b0 - ((s0 + s1) + (s2 + s3))) / Hr[r];
    }
    // backward: L^T x = y
    for (int r = N_SUP - 1; r >= 0; --r) {
      float s0 = 0.f, s1 = 0.f, s2 = 0.f, s3 = 0.f;
      int q = r + 1;
      for (; q + 3 < N_SUP; q += 4) {
        s0 += H[q * HLD + r]       * col[q * cs];
        s1 += H[(q + 1) * HLD + r] * col[(q + 1) * cs];
        s2 += H[(q + 2) * HLD + r] * col[(q + 2) * cs];
        s3 += H[(q + 3) * HLD + r] * col[(q + 3) * cs];
      }
      for (; q < N_SUP; ++q) s0 += H[q * HLD + r] * col[q * cs];
      col[r * cs] = (col[r * cs] - ((s0 + s1) + (s2 + s3))) / H[r * HLD + r];
    }
  }
}

// =====================================================================
// Schur: S = sum_a H_a^-1 ; rhs = A@Hinv_r1 + rp ; chol(S); solve dnu.
// =====================================================================
__global__ __launch_bounds__(512)
void k_schur(float* __restrict__ ws) {
  const int tid = threadIdx.x, nt = 512;
  const float* G = ws + OFF_G;
  const float* hr1 = ws + OFF_HR1; const float* rp = ws + OFF_RP;
  float* Smat = ws + OFF_SMAT; float* rhs = ws + OFF_RHS;
  float* dnu = ws + OFF_DNU;

  for (int idx = tid; idx < N_SUP * N_SUP; idx += nt) {
    int i = idx / N_SUP, j = idx % N_SUP;
    float acc = 0.f;
    for (int a = 0; a < N_WAY; ++a) acc += G[a * MAT_STRIDE + i * HLD + j];
    Smat[i * HLD + j] = acc;
  }
  for (int i = tid; i < N_SUP; i += nt) {
    float acc = rp[i];
    for (int a = 0; a < N_WAY; ++a) acc += hr1[i * N_WAY + a];
    rhs[i] = acc;
  }
  __syncthreads();
  block_cholesky(Smat, N_SUP, HLD, tid, nt);
  if (tid == 0) {
    for (int r = 0; r < N_SUP; ++r) {
      const float* Sr = Smat + r * HLD;
      float s0 = 0.f, s1 = 0.f, s2 = 0.f, s3 = 0.f;
      int q = 0;
      for (; q + 3 < r; q += 4) {
        s0 += Sr[q] * dnu[q];     s1 += Sr[q + 1] * dnu[q + 1];
        s2 += Sr[q + 2] * dnu[q + 2]; s3 += Sr[q + 3] * dnu[q + 3];
      }
      for (; q < r; ++q) s0 += Sr[q] * dnu[q];
      dnu[r] = (rhs[r] - ((s0 + s1) + (s2 + s3))) / Sr[r];
    }
    for (int r = N_SUP - 1; r >= 0; --r) {
      float s0 = 0.f, s1 = 0.f, s2 = 0.f, s3 = 0.f;
      int q = r + 1;
      for (; q + 3 < N_SUP; q += 4) {
        s0 += Smat[q * HLD + r] * dnu[q];
        s1 += Smat[(q + 1) * HLD + r] * dnu[q + 1];
        s2 += Smat[(q + 2) * HLD + r] * dnu[q + 2];
        s3 += Smat[(q + 3) * HLD + r] * dnu[q + 3];
      }
      for (; q < N_SUP; ++q) s0 += Smat[q * HLD + r] * dnu[q];
      dnu[r] = (dnu[r] - ((s0 + s1) + (s2 + s3))) / Smat[r * HLD + r];
    }
  }
}

// =====================================================================
// dz = Hinv_r1 - H^-1 dnu ; dlam = (lam*dz - rc)/s ; atomic min of ratios
// (all candidate ratios are > 0, so uint bit-pattern ordering is valid).
// =====================================================================
__global__ __launch_bounds__(256)
void k_step1(float* __restrict__ ws) {
  const int t = blockIdx.x * blockDim.x + threadIdx.x;
  if (t >= NTOT) return;
  const float* G = ws + OFF_G; const float* hr1 = ws + OFF_HR1;
  const float* dnu = ws + OFF_DNU; const float* rc = ws + OFF_RC;
  const float* lv = ws + OFF_LAM; const float* sv = ws + OFF_S;
  float* dz = ws + OFF_DZ; float* dlam = ws + OFF_DLAM;

  int j = t / N_WAY, a = t % N_WAY;
  const float* Gr = G + a * MAT_STRIDE + j * HLD;
  float s0 = 0.f, s1 = 0.f, s2 = 0.f, s3 = 0.f;
  for (int i = 0; i < N_SUP; i += 2) {             // 150 even
    s0 += Gr[i] * dnu[i];
    s1 += Gr[i + 1] * dnu[i + 1];
  }
  float acc = hr1[t] - ((s0 + s1) + (s2 + s3));
  dz[t] = acc;
  float dl = (lv[t] * acc - rc[t]) / sv[t];
  dlam[t] = dl;
  float amin = 1e30f;
  float dst = -acc;                                // ds
  if (dst < 0.f) amin = fminf(amin, -sv[t] / dst);
  if (dl  < 0.f) amin = fminf(amin, -lv[t] / dl);
  if (amin < 1e30f)
    atomicMin((unsigned int*)(ws + OFF_ALPHA), __float_as_uint(amin));
}

// =====================================================================
// alpha = min(1, 0.99*minratio); update z, s, lam, nu.
// =====================================================================
__global__ __launch_bounds__(512)
void k_step2(float* __restrict__ ws) {
  const int tid = threadIdx.x, nt = 512;
  float rmin = __uint_as_float(((unsigned int*)(ws + OFF_ALPHA))[0]);
  float alpha = fminf(1.f, 0.99f * rmin);
  float* zv = ws + OFF_Z; float* sv = ws + OFF_S; float* lv = ws + OFF_LAM;
  const float* dz = ws + OFF_DZ; const float* dlam = ws + OFF_DLAM;
  for (int t = tid; t < NTOT; t += nt) {
    zv[t] += alpha * dz[t];
    sv[t] -= alpha * dz[t];
    lv[t] += alpha * dlam[t];
  }
  for (int i = tid; i < N_SUP; i += nt)
    (ws + OFF_NU)[i] += alpha * (ws + OFF_DNU)[i];
}

// =====================================================================
// W = S^T z  (1024 x 10, padded to 16 cols); also zero the loss slot.
// =====================================================================
__global__ void k_wmat(const float* __restrict__ feat, const float* __restrict__ zv,
                       float* __restrict__ W, float* __restrict__ out) {
  int idx = blockIdx.x * blockDim.x + threadIdx.x;
  if (idx == 0) out[N_QUERY * N_WAY] = 0.f;
  if (idx >= D_FEAT * 16) return;
  int d = idx >> 4, a = idx & 15;
  float acc = 0.f;
  if (a < N_WAY)
    for (int s = 0; s < N_SUP; ++s)
      acc += feat[(size_t)s * D_FEAT + d] * zv[s * N_WAY + a];
  W[d * 16 + a] = acc;
}

// =====================================================================
// logits = scale * Q @ W (single streaming pass over the 32MB query set),
// fused log-softmax + mean NLL. One wave per 16-row tile.
// =====================================================================
__global__ __launch_bounds__(32)
void k_logits(const float* __restrict__ feat, const float* __restrict__ W,
              const float* __restrict__ scale, const int* __restrict__ lbl_q,
              float* __restrict__ out) {
  __shared__ float tile[16 * 16];
  const int lane = threadIdx.x;
  const int mh = lane & 15, hi = lane >> 4;
  const int m0 = blockIdx.x * 16;
  const float* Ar = feat + (size_t)(N_SUP + m0 + mh) * D_FEAT;
  v8f acc = {};
  for (int k0 = 0; k0 < D_FEAT; k0 += 4) {
    v2f a, b;
    a.x = Ar[k0 + 2 * hi];              a.y = Ar[k0 + 2 * hi + 1];
    b.x = W[(k0 + 2 * hi) * 16 + mh];   b.y = W[(k0 + 2 * hi + 1) * 16 + mh];
    acc = __builtin_amdgcn_wmma_f32_16x16x4_f32(false, a, false, b,
                                                (short)0, acc, false, false);
  }
  float scl = scale[0];
  for (int v = 0; v < 8; ++v)
    tile[(v + 8 * hi) * 16 + mh] = acc[v] * scl;
  __syncthreads();
  if (lane < 16) {
    int row = m0 + lane;
    float maxv = -1e30f;
    for (int c = 0; c < N_WAY; ++c) maxv = fmaxf(maxv, tile[lane * 16 + c]);
    float sum = 0.f;
    for (int c = 0; c < N_WAY; ++c) sum += __expf(tile[lane * 16 + c] - maxv);
    float lse = maxv + __logf(sum);
    for (int c = 0; c < N_WAY; ++c)
      out[row * N_WAY + c] = tile[lane * 16 + c] - lse;
    int lbl = lbl_q[row];
    atomicAdd(&out[N_QUERY * N_WAY], -(tile[lane * 16 + lbl] - lse) / (float)N_QUERY);
  }
}

// =====================================================================
extern "C" void kernel_launch(void* const* d_in, const int* in_sizes, int n_in,
                              void* d_out, int out_size, void* d_ws, size_t ws_size,
                              hipStream_t stream) {
  const float* feat    = (const float*)d_in[0];
  const int*   lbl_sup = (const int*)d_in[1];
  const int*   lbl_q   = (const int*)d_in[2];
  const float* scale   = (const float*)d_in[3];
  float* out = (float*)d_out;
  float* ws  = (float*)d_ws;   // needs ~2.14 MB

  k_gram   <<<dim3(10, 10), 32, 0, stream>>>(feat, ws + OFF_K);
  k_qp_init<<<1, 512, 0, stream>>>(ws, lbl_sup);
  for (int iter = 0; iter < MAX_ITER; ++iter) {
    k_resid<<<1, 512, 0, stream>>>(ws, lbl_sup);
    k_way  <<<N_WAY, 256, 0, stream>>>(ws);
    k_schur<<<1, 512, 0, stream>>>(ws);
    k_step1<<<(NTOT + 255) / 256, 256, 0, stream>>>(ws);
    k_step2<<<1, 512, 0, stream>>>(ws);
  }
  k_wmat  <<<(D_FEAT * 16 + 255) / 256, 256, 0, stream>>>(feat, ws + OFF_Z,
                                                          ws + OFF_W, out);
  k_logits<<<N_QUERY / 16, 32, 0, stream>>>(feat, ws + OFF_W, scale, lbl_q, out);
}